// FlashHead_48275432407819
// MI455X (gfx1250) — compile-verified
//
#include <hip/hip_runtime.h>
#include <hip/hip_bf16.h>
#include <math.h>

#define D_DIM      2048
#define NCLUST     1024
#define NPROBES    64
#define MAPLEN     256
#define KTOT       (NPROBES * MAPLEN)   // 16384 gathered rows

typedef float v2f __attribute__((ext_vector_type(2)));
typedef float v8f __attribute__((ext_vector_type(8)));

// ---------------------------------------------------------------------------
// Kernel 1: sims[c] = (h . centroids[:,c]) / ||centroids[:,c]||
// centroids laid out [D, C] row-major -> column c is stride-NCLUST; we make
// threads contiguous in c so every d-step is a fully coalesced 1 KB read.
// ---------------------------------------------------------------------------
__global__ void __launch_bounds__(256)
fh_sims(const float* __restrict__ h,
        const float* __restrict__ cent,
        float* __restrict__ sims) {
    __shared__ float s_h[D_DIM];
    const int tx = threadIdx.x;
    for (int i = tx; i < D_DIM; i += 256) s_h[i] = h[i];
    __syncthreads();

    const int c = blockIdx.x * 256 + tx;
    float dot = 0.f, nrm = 0.f;
    #pragma unroll 4
    for (int d = 0; d < D_DIM; ++d) {
        const float v = cent[(size_t)d * NCLUST + c];
        dot = fmaf(s_h[d], v, dot);
        nrm = fmaf(v, v, nrm);
    }
    sims[c] = dot / sqrtf(nrm);
}

// ---------------------------------------------------------------------------
// Kernel 2: exact top-64 by rank counting. rank(c) = #{c' : v[c']>v[c] or
// (v[c']==v[c] and c'<c)} reproduces lax.top_k descending order with
// lower-index tie-break. Single barrier, LDS broadcast reads.
// ---------------------------------------------------------------------------
__global__ void __launch_bounds__(1024)
fh_topk(const float* __restrict__ sims, int* __restrict__ topc) {
    __shared__ float s[NCLUST];
    const int tx = threadIdx.x;
    s[tx] = sims[tx];
    __syncthreads();
    const float v = s[tx];
    int r = 0;
    for (int k = 0; k < NCLUST; ++k) {
        const float vk = s[k];
        r += (vk > v) || ((vk == v) && (k < tx));
    }
    if (r < NPROBES) topc[r] = tx;
}

// ---------------------------------------------------------------------------
// Kernel 3: gathered-row matvec via V_WMMA_F32_16X16X4_F32.
// Each wave owns 16 rows (M = lane%16). ISA A layout (16x4 f32): lanes 0-15
// hold K={0,1} in v0/v1, lanes 16-31 hold K={2,3} -> one float2 load per lane
// per WMMA. B (4x16) is h replicated across all 16 columns: same {kk,kk+1}
// pair per lane, served from LDS. All accumulator columns equal the 16 dots.
// ---------------------------------------------------------------------------
__global__ void __launch_bounds__(256)
fh_logits(const float* __restrict__ h,
          const float* __restrict__ W,
          const int* __restrict__ topc,
          const int* __restrict__ vmap,
          float* __restrict__ logits) {
    __shared__ float s_h[D_DIM];
    const int tx = threadIdx.x;
    for (int i = tx * 4; i < D_DIM; i += 256 * 4)
        *(float4*)&s_h[i] = *(const float4*)&h[i];
    __syncthreads();

    const int wave    = tx >> 5;
    const int lane    = tx & 31;
    const int rowBase = blockIdx.x * 128 + wave * 16;   // 8 waves * 16 rows
    const int m       = lane & 15;
    const int row     = rowBase + m;

    const int cl  = topc[row >> 8];
    const int tok = vmap[cl * MAPLEN + (row & 255)];

    const int khalf = (lane >= 16) ? 2 : 0;
    const float* __restrict__ wp = W + (size_t)tok * D_DIM + khalf;
    const float* __restrict__ hp = s_h + khalf;

    v8f acc = (v8f)(0.0f);
    #pragma unroll 8
    for (int k = 0; k < D_DIM; k += 4) {
        v2f a = *(const v2f*)(wp + k);   // A[m][kk], A[m][kk+1]
        v2f b = *(const v2f*)(hp + k);   // B[kk][n]=h[kk] replicated over n
        acc = __builtin_amdgcn_wmma_f32_16x16x4_f32(
            /*neg_a=*/false, a, /*neg_b=*/false, b,
            /*c_mod=*/(short)0, acc, /*reuse_a=*/false, /*reuse_b=*/false);
    }

    // C/D layout: VGPR j, lanes 0-15 -> M=j (N=lane); lanes 16-31 -> M=j+8.
    // Column N=0 lives in lane 0 (rows 0-7) and lane 16 (rows 8-15).
    if (lane == 0) {
        #pragma unroll
        for (int j = 0; j < 8; ++j) logits[rowBase + j] = acc[j];
    } else if (lane == 16) {
        #pragma unroll
        for (int j = 0; j < 8; ++j) logits[rowBase + 8 + j] = acc[j];
    }
}

// ---------------------------------------------------------------------------
// Kernel 4: argmax over 16384 logits (first-occurrence tie-break, matching
// jnp.argmax), then map back through topc/vocab_maps to the token id.
// ---------------------------------------------------------------------------
__global__ void __launch_bounds__(256)
fh_argmax(const float* __restrict__ logits,
          const int* __restrict__ topc,
          const int* __restrict__ vmap,
          float* __restrict__ out_tok) {
    __shared__ float bv[256];
    __shared__ int   bi[256];
    const int tx = threadIdx.x;
    float best = -INFINITY;
    int   bidx = 0x7fffffff;
    for (int i = tx; i < KTOT; i += 256) {
        const float v = logits[i];
        if (v > best) { best = v; bidx = i; }   // ascending i -> first max kept
    }
    bv[tx] = best; bi[tx] = bidx;
    __syncthreads();
    for (int s = 128; s > 0; s >>= 1) {
        if (tx < s) {
            const float v2 = bv[tx + s];
            const int   i2 = bi[tx + s];
            if (v2 > bv[tx] || (v2 == bv[tx] && i2 < bi[tx])) { bv[tx] = v2; bi[tx] = i2; }
        }
        __syncthreads();
    }
    if (tx == 0) {
        const int r   = bi[0];
        const int cl  = topc[r >> 8];
        const int tok = vmap[cl * MAPLEN + (r & 255)];
        out_tok[0] = (float)tok;
    }
}

// ---------------------------------------------------------------------------
// Host-side launcher. Inputs (setup_inputs order):
//   d_in[0] hidden_states  f32 [1,1,2048]
//   d_in[1] lm_head_weight f32 [128000,2048]
//   d_in[2] centroids      f32 [2048,1024]
//   d_in[3] vocab_maps     i32 [1024,256]
// d_out: [ token(1) | logits(16384) ] as float.
// ---------------------------------------------------------------------------
extern "C" void kernel_launch(void* const* d_in, const int* in_sizes, int n_in,
                              void* d_out, int out_size, void* d_ws, size_t ws_size,
                              hipStream_t stream) {
    const float* h    = (const float*)d_in[0];
    const float* W    = (const float*)d_in[1];
    const float* cent = (const float*)d_in[2];
    const int*   vmap = (const int*)d_in[3];

    float* sims = (float*)d_ws;                 // 1024 floats
    int*   topc = (int*)(sims + NCLUST);        // 64 ints

    float* out    = (float*)d_out;
    float* logits = out + 1;

    fh_sims  <<<NCLUST / 256, 256, 0, stream>>>(h, cent, sims);
    fh_topk  <<<1, NCLUST,      0, stream>>>(sims, topc);
    fh_logits<<<KTOT / 128, 256, 0, stream>>>(h, W, topc, vmap, logits);
    fh_argmax<<<1, 256,          0, stream>>>(logits, topc, vmap, out);
}